// SparseProjector_86938728006311
// MI455X (gfx1250) — compile-verified
//
#include <hip/hip_runtime.h>

// Problem constants (from reference): B=2, N_SRC=N_DST=131072, E=524288, F=64
#define NB      2
#define NSRC    131072
#define NDST    131072
#define NEDGE   524288
#define FDIM    64
#define EPSV    1e-8f

typedef __attribute__((ext_vector_type(2))) float v2f;
typedef __attribute__((ext_vector_type(8))) float v8f;

// ---------------- workspace layout (4-byte element offsets) ----------------
#define WS_COUNTS   0                        // int[NDST+1]: counts, then CSR offsets after scan
#define WS_CURSOR   (WS_COUNTS + NDST + 1)   // int[NDST]  : scatter cursors
#define WS_NORM     (WS_CURSOR + NDST)       // float[NDST]: per-dst weight sums
#define WS_BSUMS    (WS_NORM + NDST)         // int[512]   : scan block sums
#define WS_SRCP     (WS_BSUMS + 512)         // int[NEDGE] : sorted (src | dst_local<<24)
#define WS_WSORT    (WS_SRCP + NEDGE)        // float[NEDGE]: sorted normalized weights
// total ~5.8 MB

__global__ void k_zero(int* __restrict__ ws) {
  int i = blockIdx.x * blockDim.x + threadIdx.x;
  if (i < WS_SRCP) ws[i] = 0;
}

__global__ void k_count(const int* __restrict__ dst, const float* __restrict__ w,
                        int* __restrict__ counts, float* __restrict__ norm) {
  int e = blockIdx.x * blockDim.x + threadIdx.x;
  if (e < NEDGE) {
    int d = dst[e];
    atomicAdd(&counts[d], 1);
    atomicAdd(&norm[d], w[e]);
  }
}

// per-block reduction of counts -> block sums (512 blocks x 256)
__global__ void k_scanA(const int* __restrict__ counts, int* __restrict__ bsums) {
  __shared__ int sh[256];
  int t = threadIdx.x;
  sh[t] = counts[blockIdx.x * 256 + t];
  __syncthreads();
  for (int off = 128; off > 0; off >>= 1) {
    if (t < off) sh[t] += sh[t + off];
    __syncthreads();
  }
  if (t == 0) bsums[blockIdx.x] = sh[0];
}

// single-block exclusive scan of 512 block sums; also writes offsets[NDST]=E
__global__ void k_scanB(int* __restrict__ bsums, int* __restrict__ offsets) {
  __shared__ int sh[512];
  int t = threadIdx.x;
  sh[t] = bsums[t];
  __syncthreads();
  for (int off = 1; off < 512; off <<= 1) {
    int v = (t >= off) ? sh[t - off] : 0;
    __syncthreads();
    sh[t] += v;
    __syncthreads();
  }
  bsums[t] = (t == 0) ? 0 : sh[t - 1];
  if (t == 511) offsets[NDST] = sh[511];
}

// per-block exclusive scan + block offset, in place (counts -> CSR offsets)
__global__ void k_scanC(int* __restrict__ cnt_off, const int* __restrict__ bsums) {
  __shared__ int sh[256];
  int t = threadIdx.x;
  int i = blockIdx.x * 256 + t;
  int c = cnt_off[i];
  sh[t] = c;
  __syncthreads();
  for (int off = 1; off < 256; off <<= 1) {
    int v = (t >= off) ? sh[t - off] : 0;
    __syncthreads();
    sh[t] += v;
    __syncthreads();
  }
  cnt_off[i] = (sh[t] - c) + bsums[blockIdx.x];
}

// scatter edges into CSR order; store normalized weight and src packed with dst&15
__global__ void k_scatter(const int* __restrict__ src, const int* __restrict__ dst,
                          const float* __restrict__ w, const int* __restrict__ offsets,
                          int* __restrict__ cursor, const float* __restrict__ norm,
                          int* __restrict__ srcp, float* __restrict__ wsort) {
  int e = blockIdx.x * blockDim.x + threadIdx.x;
  if (e < NEDGE) {
    int d = dst[e];
    int pos = offsets[d] + atomicAdd(&cursor[d], 1);
    srcp[pos]  = (src[e] & 0x00FFFFFF) | ((d & 15) << 24);
    wsort[pos] = w[e] / (norm[d] + EPSV);
  }
}

// WMMA SpMM: one wave handles 16 destinations; chunks of 4 CSR edges form
// A(16x4 selection*weight) x B(4x16 gathered x-tile) -> 16x16 f32 acc, 4 tiles for F=64.
// Loop bounds forced into SGPRs (readfirstlane): scalar loop control, statically
// uniform EXEC at the WMMAs, and edge metadata becomes SMEM (s_load) traffic.
__global__ void __launch_bounds__(256)
k_spmm(const float* __restrict__ x, const int* __restrict__ offsets,
       const int* __restrict__ srcp, const float* __restrict__ wsort,
       float* __restrict__ out) {
  const int lane = threadIdx.x & 31;
  const int wave = threadIdx.x >> 5;
  const int dstBlock = blockIdx.x * 8 + wave;   // 16 destinations per wave
  const int b = blockIdx.y;
  const int dBase = dstBlock << 4;
  // wave-uniform CSR edge range -> SGPRs
  const int eBeg = __builtin_amdgcn_readfirstlane(offsets[dBase]);
  const int eEnd = __builtin_amdgcn_readfirstlane(offsets[dBase + 16]);

  const int m  = lane & 15;   // A row (M) / B col (N)
  const int hi = lane >> 4;   // half-wave select: A/B K-slots {0,1} vs {2,3}

  v8f acc0 = {}, acc1 = {}, acc2 = {}, acc3 = {};
  const float* xb = x + (size_t)b * NSRC * FDIM;

  for (int p = eBeg; p < eEnd; p += 4) {
    // 4 edges' metadata: uniform addresses -> scalar loads; tail handled by
    // clamped index + zeroed weight (zero A column => B row is don't-care).
    int   pkv[4];
    float wvv[4];
    #pragma unroll
    for (int k = 0; k < 4; ++k) {
      int  q  = p + k;
      bool in = q < eEnd;          // uniform
      int  qs = in ? q : eBeg;     // uniform, always valid
      pkv[k] = srcp[qs];
      float wk = wsort[qs];
      wvv[k] = in ? wk : 0.0f;
    }

    // per-lane K-slot selection (A 16x4 ISA layout: V0 K={0|2}, V1 K={1|3})
    int   pk0 = hi ? pkv[2] : pkv[0];
    int   pk1 = hi ? pkv[3] : pkv[1];
    float w0  = hi ? wvv[2] : wvv[0];
    float w1  = hi ? wvv[3] : wvv[1];
    int dl0 = (pk0 >> 24) & 15, dl1 = (pk1 >> 24) & 15;
    const float* r0 = xb + (size_t)(pk0 & 0x00FFFFFF) * FDIM;
    const float* r1 = xb + (size_t)(pk1 & 0x00FFFFFF) * FDIM;

    v2f a;
    a.x = (dl0 == m) ? w0 : 0.0f;
    a.y = (dl1 == m) ? w1 : 0.0f;

    v2f b0, b1, b2, b3;
    b0.x = r0[m];      b0.y = r1[m];
    b1.x = r0[16 + m]; b1.y = r1[16 + m];
    b2.x = r0[32 + m]; b2.y = r1[32 + m];
    b3.x = r0[48 + m]; b3.y = r1[48 + m];

    acc0 = __builtin_amdgcn_wmma_f32_16x16x4_f32(false, a, false, b0, (short)0, acc0, false, false);
    acc1 = __builtin_amdgcn_wmma_f32_16x16x4_f32(false, a, false, b1, (short)0, acc1, false, false);
    acc2 = __builtin_amdgcn_wmma_f32_16x16x4_f32(false, a, false, b2, (short)0, acc2, false, false);
    acc3 = __builtin_amdgcn_wmma_f32_16x16x4_f32(false, a, false, b3, (short)0, acc3, false, false);
  }

  // C/D layout: VGPR r -> lanes0-15: M=r, lanes16-31: M=r+8; N=lane&15
  float* ob = out + ((size_t)b * NDST + dBase) * FDIM;
  #pragma unroll
  for (int r = 0; r < 8; ++r) {
    size_t row = (size_t)(r + hi * 8) * FDIM;
    ob[row + m]      = acc0[r];
    ob[row + 16 + m] = acc1[r];
    ob[row + 32 + m] = acc2[r];
    ob[row + 48 + m] = acc3[r];
  }
}

extern "C" void kernel_launch(void* const* d_in, const int* in_sizes, int n_in,
                              void* d_out, int out_size, void* d_ws, size_t ws_size,
                              hipStream_t stream) {
  const float* x   = (const float*)d_in[0];
  const int*   ei  = (const int*)d_in[1];   // edge_index flat [2][E] (int32 per harness)
  const float* wts = (const float*)d_in[2];
  float* out = (float*)d_out;
  int*   ws  = (int*)d_ws;

  const int* src = ei;
  const int* dst = ei + NEDGE;
  int*   counts = ws + WS_COUNTS;           // becomes CSR offsets after scan
  int*   cursor = ws + WS_CURSOR;
  float* norm   = (float*)(ws + WS_NORM);
  int*   bsums  = ws + WS_BSUMS;
  int*   srcp   = ws + WS_SRCP;
  float* wsort  = (float*)(ws + WS_WSORT);

  k_zero   <<<(WS_SRCP + 255) / 256, 256, 0, stream>>>(ws);
  k_count  <<<NEDGE / 256, 256, 0, stream>>>(dst, wts, counts, norm);
  k_scanA  <<<NDST / 256, 256, 0, stream>>>(counts, bsums);
  k_scanB  <<<1, 512, 0, stream>>>(bsums, counts);
  k_scanC  <<<NDST / 256, 256, 0, stream>>>(counts, bsums);
  k_scatter<<<NEDGE / 256, 256, 0, stream>>>(src, dst, wts, counts, cursor, norm, srcp, wsort);
  k_spmm   <<<dim3(NDST / 16 / 8, NB), 256, 0, stream>>>(x, counts, srcp, wsort, out);
}